// GGNN_NODE_17952963297399
// MI455X (gfx1250) — compile-verified
//
#include <hip/hip_runtime.h>
#include <hip/hip_bf16.h>
#include <math.h>

// ---------------- problem constants (match reference) ----------------
#define N_NODES   100000
#define N_LAYERS  5
#define HID       150
#define HP        160            // padded hidden (10 tiles of 16)
#define RPB       32             // rows per block (2 row tiles)
#define RBLOCKS   (N_NODES / RPB) // 3125, exact
#define COL_TILES (HP / 16)      // 10
#define KSTEPS    (HP / 4)       // 40 wmma k-steps of 4
#define LDS_LD    164            // LDS row stride in floats (bank-conflict free)

typedef __attribute__((ext_vector_type(2))) float v2f;
typedef __attribute__((ext_vector_type(8))) float v8f;

__device__ __forceinline__ v8f wmma_f32(v2f a, v2f b, v8f c) {
  // V_WMMA_F32_16X16X4_F32 : D(16x16,f32) = A(16x4,f32) x B(4x16,f32) + C
  return __builtin_amdgcn_wmma_f32_16x16x4_f32(
      /*neg_a=*/false, a, /*neg_b=*/false, b,
      /*c_mod=*/(short)0, c, /*reuse_a=*/false, /*reuse_b=*/false);
}

__device__ __forceinline__ float sigmoidf_(float x) {
  return 1.0f / (1.0f + __expf(-x));
}

// ---------------- prep: pad h, pack weights into B-fragment order ----------
// B-frag layout (mirrors documented A 16x4 layout): for lane l, vgpr j:
//   B[k = j + 2*(l>=16)][n = l%16]

__global__ void init_hpad(const float* __restrict__ x, float* __restrict__ hpad) {
  int idx = blockIdx.x * blockDim.x + threadIdx.x;
  if (idx >= N_NODES * HP) return;
  int row = idx / HP, col = idx - row * HP;
  hpad[idx] = (col < HID) ? x[row * HID + col] : 0.0f;
}

// weight[l][k][n]  (h @ weight[l])  ->  wpack1[l][ct][ks][lane][j]
__global__ void pack_w1(const float* __restrict__ w, float* __restrict__ wp) {
  int idx = blockIdx.x * blockDim.x + threadIdx.x;
  const int per_layer = COL_TILES * KSTEPS * 64;
  if (idx >= N_LAYERS * per_layer) return;
  int l    = idx / per_layer;
  int rem  = idx - l * per_layer;
  int j    = rem & 1;
  int lane = (rem >> 1) & 31;
  int ks   = (rem >> 6) % KSTEPS;
  int ct   = (rem >> 6) / KSTEPS;
  int k = ks * 4 + 2 * (lane >> 4) + j;
  int n = ct * 16 + (lane & 15);
  wp[idx] = (k < HID && n < HID) ? w[(l * HID + k) * HID + n] : 0.0f;
}

// w_ih / w_hh are [3H, H]; B = W^T so B[k][n_glob] = W[n_glob][k], n_glob=g*HID+n
__global__ void pack_wg(const float* __restrict__ w, float* __restrict__ wp) {
  int idx = blockIdx.x * blockDim.x + threadIdx.x;
  const int per_gate = COL_TILES * KSTEPS * 64;
  if (idx >= 3 * per_gate) return;
  int g    = idx / per_gate;
  int rem  = idx - g * per_gate;
  int j    = rem & 1;
  int lane = (rem >> 1) & 31;
  int ks   = (rem >> 6) % KSTEPS;
  int ct   = (rem >> 6) / KSTEPS;
  int k = ks * 4 + 2 * (lane >> 4) + j;
  int n = ct * 16 + (lane & 15);
  wp[idx] = (k < HID && n < HID) ? w[(g * HID + n) * HID + k] : 0.0f;
}

// ---------------- CSR build (once per call; replaces 1.2B float atomics) ---
__global__ void hist_deg(const int* __restrict__ ei, int* __restrict__ deg,
                         int n_edges) {
  int e = blockIdx.x * blockDim.x + threadIdx.x;
  if (e >= n_edges) return;
  atomicAdd(&deg[ei[n_edges + e]], 1);   // dst
}

// single-workgroup exclusive scan over N_NODES degrees -> rowptr, cursor
__global__ void __launch_bounds__(1024)
scan_deg(const int* __restrict__ deg, int* __restrict__ rowptr,
         int* __restrict__ cursor) {
  __shared__ int buf[1024];
  __shared__ int carry_s;
  const int tid = threadIdx.x;
  if (tid == 0) carry_s = 0;
  __syncthreads();
  for (int base = 0; base < N_NODES; base += 1024) {
    int i = base + tid;
    int v = (i < N_NODES) ? deg[i] : 0;
    buf[tid] = v;
    __syncthreads();
    for (int off = 1; off < 1024; off <<= 1) {    // Hillis-Steele inclusive
      int t = (tid >= off) ? buf[tid - off] : 0;
      __syncthreads();
      buf[tid] += t;
      __syncthreads();
    }
    int incl = buf[tid];
    int carry = carry_s;
    if (i < N_NODES) {
      int excl = carry + incl - v;
      rowptr[i] = excl;
      cursor[i] = excl;
    }
    __syncthreads();
    if (tid == 1023) carry_s = carry + incl;
    __syncthreads();
  }
  if (tid == 0) rowptr[N_NODES] = carry_s;
}

__global__ void fill_csr(const int* __restrict__ ei, int* __restrict__ cursor,
                         int* __restrict__ csr_src, int n_edges) {
  int e = blockIdx.x * blockDim.x + threadIdx.x;
  if (e >= n_edges) return;
  int pos = atomicAdd(&cursor[ei[n_edges + e]], 1);
  csr_src[pos] = ei[e];                  // src
}

// ---------------- GEMM: m = h_pad @ W[l]  (100000x160 * 160x160) ----------
// block = 320 thr = 10 waves, 32 rows (2 row tiles); wave w -> col tile w.
// Each B-fragment load feeds both row tiles (halves L2 weight traffic).
__global__ void __launch_bounds__(320)
gemm_m(const float* __restrict__ h, const float* __restrict__ wp,
       float* __restrict__ m) {
  __shared__ float hs[RPB * LDS_LD];
  const int rowbase = blockIdx.x * RPB;

  for (int i = threadIdx.x; i < RPB * (HP / 4); i += 320) {
    int r = i / (HP / 4), cc = (i - r * (HP / 4)) * 4;
    float4 v = *(const float4*)(h + (rowbase + r) * HP + cc);
    float* d = &hs[r * LDS_LD + cc];
    d[0] = v.x; d[1] = v.y; d[2] = v.z; d[3] = v.w;
  }
  __syncthreads();

  const int wave = threadIdx.x >> 5, lane = threadIdx.x & 31;
  const int half = lane >> 4, l16 = lane & 15;
  const float* wpt = wp + wave * (KSTEPS * 64);

  v8f acc0 = {}, acc1 = {};
  for (int ks = 0; ks < KSTEPS; ++ks) {
    int ao = ks * 4 + 2 * half;
    v2f a0 = *(const v2f*)&hs[l16 * LDS_LD + ao];
    v2f a1 = *(const v2f*)&hs[(l16 + 16) * LDS_LD + ao];
    v2f b  = *(const v2f*)(wpt + ks * 64 + lane * 2);
    acc0 = wmma_f32(a0, b, acc0);
    acc1 = wmma_f32(a1, b, acc1);
  }
  const int col = wave * 16 + l16;
  #pragma unroll
  for (int v = 0; v < 8; ++v) {
    m[(rowbase + v + 8 * half) * HP + col]      = acc0[v];
    m[(rowbase + 16 + v + 8 * half) * HP + col] = acc1[v];
  }
}

// ---------------- pull-style aggregation: aggr[i] = sum m[src] ------------
// one wave per destination node; b128 + b32 per edge (2 vmem instrs);
// rows are 640B-aligned so 16B vector loads are legal. No float atomics.
__global__ void __launch_bounds__(256)
gather_aggr(const float* __restrict__ m, const int* __restrict__ rowptr,
            const int* __restrict__ csr_src, float* __restrict__ aggr) {
  int node = blockIdx.x * 8 + (threadIdx.x >> 5);
  if (node >= N_NODES) return;
  const int lane = threadIdx.x & 31;
  int beg = rowptr[node], end = rowptr[node + 1];
  float a0 = 0.f, a1 = 0.f, a2 = 0.f, a3 = 0.f, a4 = 0.f;
  for (int j = beg; j < end; ++j) {
    const float* mr = m + (size_t)csr_src[j] * HP;
    float4 v = *(const float4*)(mr + 4 * lane);   // cols 0..127
    float  s = mr[128 + lane];                    // cols 128..159
    a0 += v.x; a1 += v.y; a2 += v.z; a3 += v.w; a4 += s;
  }
  float* ar = aggr + (size_t)node * HP;
  float4 o; o.x = a0; o.y = a1; o.z = a2; o.w = a3;
  *(float4*)(ar + 4 * lane) = o;
  ar[128 + lane] = a4;
}

// ---------------- fused GRU cell ------------------------------------------
// block = 320 thr = 10 waves, 32 rows; wave w owns 16-col tile w of each gate.
// 12 tile-GEMMs per wave (2 row tiles x {ir,hr,iz,hz,in,hn}); every B-frag
// is reused across both row tiles. Gates + blend in registers; in-place h.
__global__ void __launch_bounds__(320)
gru_fused(const float* __restrict__ hin, const float* __restrict__ aggr,
          const float* __restrict__ wih, const float* __restrict__ whh,
          const float* __restrict__ bih, const float* __restrict__ bhh,
          float* __restrict__ hout, int ldo, int zero_pad) {
  __shared__ float hs[RPB * LDS_LD];
  __shared__ float as_[RPB * LDS_LD];
  const int rowbase = blockIdx.x * RPB;

  for (int i = threadIdx.x; i < 2 * RPB * (HP / 4); i += 320) {
    int which = i / (RPB * (HP / 4));
    int t = i - which * (RPB * (HP / 4));
    int r = t / (HP / 4), cc = (t - r * (HP / 4)) * 4;
    const float* srcp = which ? (aggr + (rowbase + r) * HP + cc)
                              : (hin + (rowbase + r) * HP + cc);
    float4 v = *(const float4*)srcp;
    float* d = which ? &as_[r * LDS_LD + cc] : &hs[r * LDS_LD + cc];
    d[0] = v.x; d[1] = v.y; d[2] = v.z; d[3] = v.w;
  }
  __syncthreads();

  const int wave = threadIdx.x >> 5, lane = threadIdx.x & 31;
  const int half = lane >> 4, l16 = lane & 15;
  const int ct = wave;

  const float* pir = wih + (0 * COL_TILES + ct) * (KSTEPS * 64);
  const float* piz = wih + (1 * COL_TILES + ct) * (KSTEPS * 64);
  const float* pin = wih + (2 * COL_TILES + ct) * (KSTEPS * 64);
  const float* phr = whh + (0 * COL_TILES + ct) * (KSTEPS * 64);
  const float* phz = whh + (1 * COL_TILES + ct) * (KSTEPS * 64);
  const float* phn = whh + (2 * COL_TILES + ct) * (KSTEPS * 64);

  v8f ir0 = {}, hr0 = {}, iz0 = {}, hz0 = {}, in0 = {}, hn0 = {};
  v8f ir1 = {}, hr1 = {}, iz1 = {}, hz1 = {}, in1 = {}, hn1 = {};
  for (int ks = 0; ks < KSTEPS; ++ks) {
    int ao = ks * 4 + 2 * half;
    v2f aA0 = *(const v2f*)&as_[l16 * LDS_LD + ao];
    v2f aA1 = *(const v2f*)&as_[(l16 + 16) * LDS_LD + ao];
    v2f aH0 = *(const v2f*)&hs[l16 * LDS_LD + ao];
    v2f aH1 = *(const v2f*)&hs[(l16 + 16) * LDS_LD + ao];
    v2f br = *(const v2f*)(pir + ks * 64 + lane * 2);
    v2f bz = *(const v2f*)(piz + ks * 64 + lane * 2);
    v2f bn = *(const v2f*)(pin + ks * 64 + lane * 2);
    v2f cr = *(const v2f*)(phr + ks * 64 + lane * 2);
    v2f cz = *(const v2f*)(phz + ks * 64 + lane * 2);
    v2f cn = *(const v2f*)(phn + ks * 64 + lane * 2);
    ir0 = wmma_f32(aA0, br, ir0);  ir1 = wmma_f32(aA1, br, ir1);
    iz0 = wmma_f32(aA0, bz, iz0);  iz1 = wmma_f32(aA1, bz, iz1);
    in0 = wmma_f32(aA0, bn, in0);  in1 = wmma_f32(aA1, bn, in1);
    hr0 = wmma_f32(aH0, cr, hr0);  hr1 = wmma_f32(aH1, cr, hr1);
    hz0 = wmma_f32(aH0, cz, hz0);  hz1 = wmma_f32(aH1, cz, hz1);
    hn0 = wmma_f32(aH0, cn, hn0);  hn1 = wmma_f32(aH1, cn, hn1);
  }

  const int col = ct * 16 + l16;
  const bool valid = (col < HID);
  const int cb = valid ? col : 0;
  const float bir = valid ? bih[cb]           : 0.0f;
  const float biz = valid ? bih[HID + cb]     : 0.0f;
  const float bin = valid ? bih[2 * HID + cb] : 0.0f;
  const float bhr = valid ? bhh[cb]           : 0.0f;
  const float bhz = valid ? bhh[HID + cb]     : 0.0f;
  const float bhn = valid ? bhh[2 * HID + cb] : 0.0f;

  #pragma unroll
  for (int rt = 0; rt < 2; ++rt) {
    #pragma unroll
    for (int v = 0; v < 8; ++v) {
      float irv = rt ? ir1[v] : ir0[v];
      float hrv = rt ? hr1[v] : hr0[v];
      float izv = rt ? iz1[v] : iz0[v];
      float hzv = rt ? hz1[v] : hz0[v];
      float inv = rt ? in1[v] : in0[v];
      float hnv = rt ? hn1[v] : hn0[v];
      int rl = rt * 16 + v + 8 * half;
      float r = sigmoidf_((irv + bir) + (hrv + bhr));
      float z = sigmoidf_((izv + biz) + (hzv + bhz));
      float n = tanhf((inv + bin) + r * (hnv + bhn));
      float hv = hs[rl * LDS_LD + col];
      float o = (1.0f - z) * n + z * hv;
      long row = rowbase + rl;
      if (valid)          hout[row * ldo + col] = o;
      else if (zero_pad)  hout[row * ldo + col] = 0.0f;
    }
  }
}

// ---------------- launcher ------------------------------------------------
extern "C" void kernel_launch(void* const* d_in, const int* in_sizes, int n_in,
                              void* d_out, int out_size, void* d_ws, size_t ws_size,
                              hipStream_t stream) {
  const float* x    = (const float*)d_in[0];
  const float* wgt  = (const float*)d_in[1];
  const float* w_ih = (const float*)d_in[2];
  const float* w_hh = (const float*)d_in[3];
  const float* b_ih = (const float*)d_in[4];
  const float* b_hh = (const float*)d_in[5];
  const int*   ei   = (const int*)d_in[6];
  const int n_edges = in_sizes[6] / 2;
  float* out = (float*)d_out;

  // workspace layout
  const size_t npad = (size_t)N_NODES * HP;
  const size_t gate_sz = 3 * COL_TILES * KSTEPS * 64;
  float* ws      = (float*)d_ws;
  float* h_pad   = ws;
  float* m_pad   = h_pad + npad;
  float* aggr    = m_pad + npad;
  float* wpack1  = aggr  + npad;
  float* wpackih = wpack1  + (size_t)N_LAYERS * COL_TILES * KSTEPS * 64;
  float* wpackhh = wpackih + gate_sz;
  int*   deg     = (int*)(wpackhh + gate_sz);
  int*   rowptr  = deg + N_NODES;
  int*   cursor  = rowptr + (N_NODES + 1);
  int*   csr_src = cursor + N_NODES;

  // ---- prep: pads, weight packs, CSR (deterministic work; capture-safe) ----
  {
    int n = N_NODES * HP;
    init_hpad<<<(n + 255) / 256, 256, 0, stream>>>(x, h_pad);
    int n1 = N_LAYERS * COL_TILES * KSTEPS * 64;
    pack_w1<<<(n1 + 255) / 256, 256, 0, stream>>>(wgt, wpack1);
    int ng = (int)gate_sz;
    pack_wg<<<(ng + 255) / 256, 256, 0, stream>>>(w_ih, wpackih);
    pack_wg<<<(ng + 255) / 256, 256, 0, stream>>>(w_hh, wpackhh);

    hipMemsetAsync(deg, 0, (size_t)N_NODES * sizeof(int), stream);
    hist_deg<<<(n_edges + 255) / 256, 256, 0, stream>>>(ei, deg, n_edges);
    scan_deg<<<1, 1024, 0, stream>>>(deg, rowptr, cursor);
    fill_csr<<<(n_edges + 255) / 256, 256, 0, stream>>>(ei, cursor, csr_src, n_edges);
  }

  for (int l = 0; l < N_LAYERS; ++l) {
    gemm_m<<<RBLOCKS, 320, 0, stream>>>(
        h_pad, wpack1 + (size_t)l * COL_TILES * KSTEPS * 64, m_pad);

    gather_aggr<<<(N_NODES + 7) / 8, 256, 0, stream>>>(m_pad, rowptr, csr_src, aggr);

    const bool last = (l == N_LAYERS - 1);
    gru_fused<<<RBLOCKS, 320, 0, stream>>>(
        h_pad, aggr, wpackih, wpackhh, b_ih, b_hh,
        last ? out : h_pad, last ? HID : HP, last ? 0 : 1);
  }
}